// Entmax15Attention_83786222010774
// MI455X (gfx1250) — compile-verified
//
#include <hip/hip_runtime.h>
#include <hip/hip_bf16.h>
#include <math.h>

// ---- problem constants (reference: B=4, T=1024, C=768, H=12, hd=64) ----
#define BB   4
#define TT   1024
#define CC   768
#define HH   12
#define HD   64
#define QK_SCALE 0.125f   // 1/sqrt(64)

typedef __attribute__((ext_vector_type(2))) float v2f;
typedef __attribute__((ext_vector_type(8))) float v8f;

// ---------------------------------------------------------------------------
// WMMA f32 16x16x4 fragment helpers (ISA 7.12.2 layouts, wave32):
//  A (16x4):  lane L holds row M = L%16; VGPR0 = K=(L<16?0:2), VGPR1 = K+1
//  B (4x16):  lane L holds col N = L%16; VGPR0 = K=(L<16?0:2), VGPR1 = K+1
//  C/D (16x16): lane L holds col N = L%16; VGPR r = row M = r + (L<16?0:8)
// ---------------------------------------------------------------------------
__device__ __forceinline__ v8f wmma_f32(v2f a, v2f b, v8f c) {
    return __builtin_amdgcn_wmma_f32_16x16x4_f32(false, a, false, b,
                                                 (short)0, c, false, false);
}

__device__ __forceinline__ v2f ldfrag(const float* __restrict__ p, int kk) {
    v2f f; f.x = p[kk]; f.y = p[kk + 1]; return f;
}

// ---------------------------------------------------------------------------
// Kernel 1: qkv = x @ W_attn^T + b_attn, scattered to q/k/v [B,H,T,hd]
//  M = 4096, N = 2304, K = 768.
//  Block 128 = 4 waves; each wave computes a 16(m) x 64(n) strip with
//  register double-buffering: iteration k+1 fragment loads are issued
//  before iteration k's WMMAs, so matrix ops overlap the load pipe.
//  grid (256, 9).
// ---------------------------------------------------------------------------
__global__ void qkv_gemm_kernel(const float* __restrict__ x,
                                const float* __restrict__ W,
                                const float* __restrict__ bias,
                                float* __restrict__ qo,
                                float* __restrict__ ko,
                                float* __restrict__ vo) {
    const int lane = threadIdx.x & 31;
    const int wave = threadIdx.x >> 5;
    const int m0 = blockIdx.x * 16;
    const int n0 = (blockIdx.y * 4 + wave) * 64;       // strip of 4 n-tiles

    const int l16  = lane & 15;
    const int koff = (lane < 16) ? 0 : 2;

    const float* pa = x + (long)(m0 + l16) * CC + koff;
    const float* pb[4];
#pragma unroll
    for (int j = 0; j < 4; ++j)
        pb[j] = W + (long)(n0 + j * 16 + l16) * CC + koff;

    v8f acc[4] = {};
    // software pipeline: prime iteration 0
    v2f a_cur = ldfrag(pa, 0);
    v2f b_cur[4];
#pragma unroll
    for (int j = 0; j < 4; ++j) b_cur[j] = ldfrag(pb[j], 0);

    for (int kk = 0; kk + 4 < CC; kk += 4) {
        __builtin_prefetch(pa + kk + 128, 0, 1);       // global_prefetch_b8
        // issue next-iteration loads before consuming current fragments
        v2f a_nxt = ldfrag(pa, kk + 4);
        v2f b_nxt[4];
#pragma unroll
        for (int j = 0; j < 4; ++j) b_nxt[j] = ldfrag(pb[j], kk + 4);
#pragma unroll
        for (int j = 0; j < 4; ++j) acc[j] = wmma_f32(a_cur, b_cur[j], acc[j]);
        a_cur = a_nxt;
#pragma unroll
        for (int j = 0; j < 4; ++j) b_cur[j] = b_nxt[j];
    }
#pragma unroll
    for (int j = 0; j < 4; ++j) acc[j] = wmma_f32(a_cur, b_cur[j], acc[j]);

    const int rbase = m0 + ((lane < 16) ? 0 : 8);
#pragma unroll
    for (int j = 0; j < 4; ++j) {
        const int col   = n0 + j * 16 + l16;           // n in [0, 2304)
        const int which = col / CC;                    // 0=q 1=k 2=v
        const int c     = col - which * CC;
        const int h     = c >> 6;
        const int d     = c & 63;
        const float bv  = bias[col];
        float* dst = (which == 0) ? qo : (which == 1) ? ko : vo;
#pragma unroll
        for (int r = 0; r < 8; ++r) {
            const int m    = rbase + r;                // m = b*T + t
            const int bidx = m >> 10;
            const int t    = m & (TT - 1);
            dst[((((long)bidx * HH + h) * TT) + t) * HD + d] = acc[j][r] + bv;
        }
    }
}

// ---------------------------------------------------------------------------
// Kernel 2: S[i,j] = (q_i . k_j) * scale for j <= i (lower triangle only)
//  One wave per 16(m) x 64(j) strip; Q fragment reused across 4 K tiles.
//  k-loop fully unrolled (16 steps) -> compiler schedules loads globally.
//  grid (T/16, T/64, B*H) = (64, 16, 48), block 32.
// ---------------------------------------------------------------------------
__global__ void scores_kernel(const float* __restrict__ q,
                              const float* __restrict__ k,
                              float* __restrict__ att) {
    const int lane = threadIdx.x & 31;
    const int m0 = blockIdx.x * 16;                    // query rows
    const int j0 = blockIdx.y * 64;                    // key-col strip base
    if (j0 > m0 + 15) return;                          // fully masked strip
    const int bh = blockIdx.z;

    const float* Q = q + (long)bh * TT * HD;
    const float* K = k + (long)bh * TT * HD;

    const int l16  = lane & 15;
    const int koff = (lane < 16) ? 0 : 2;

    const float* pa = Q + (long)(m0 + l16) * HD + koff;
    const float* pb[4];
#pragma unroll
    for (int j = 0; j < 4; ++j)
        pb[j] = K + (long)(j0 + j * 16 + l16) * HD + koff;

    v8f acc[4] = {};
#pragma unroll
    for (int kk = 0; kk < HD; kk += 4) {
        v2f a = ldfrag(pa, kk);
#pragma unroll
        for (int j = 0; j < 4; ++j) {
            v2f b = ldfrag(pb[j], kk);                 // (K^T)[k][j] = K[j][k]
            acc[j] = wmma_f32(a, b, acc[j]);
        }
    }

    float* S = att + (long)bh * TT * TT;
    const int rbase = m0 + ((lane < 16) ? 0 : 8);
#pragma unroll
    for (int j = 0; j < 4; ++j) {
        const int col = j0 + j * 16 + l16;
#pragma unroll
        for (int r = 0; r < 8; ++r) {
            const int row = rbase + r;
            if (col <= row)                            // causal prefix only
                S[(long)row * TT + col] = acc[j][r] * QK_SCALE;
        }
    }
}

// ---------------------------------------------------------------------------
// Kernel 3: entmax-1.5 over each causal row (in place on att).
// One 256-thread block per row (b,h,i); valid length L = i+1.
// LDS: bitonic sort (desc) + Hillis-Steele scans of v and v^2.
// ---------------------------------------------------------------------------
#define ENT_THREADS 256

__global__ __launch_bounds__(ENT_THREADS)
void entmax15_kernel(float* __restrict__ att) {
    const int tid = threadIdx.x;
    const int row = blockIdx.x;                  // flat over B*H*T
    const int i   = row & (TT - 1);
    const int L   = i + 1;
    float* z = att + (long)row * TT;

    __shared__ float s_sorted[TT];
    __shared__ float s_cs[TT];      // cumsum of sorted
    __shared__ float s_cs2[TT];     // cumsum of sorted^2
    __shared__ float s_tau[TT];
    __shared__ float s_red[ENT_THREADS];
    __shared__ int   s_cnt;
    if (tid == 0) s_cnt = 0;

    int P = 1;
    while (P < L) P <<= 1;           // pow2 padded length, <= 1024

    // ---- row max ----
    float lm = -__builtin_inff();
    for (int idx = tid; idx < L; idx += ENT_THREADS) lm = fmaxf(lm, z[idx]);
    s_red[tid] = lm;
    __syncthreads();
    for (int s = ENT_THREADS / 2; s > 0; s >>= 1) {
        if (tid < s) s_red[tid] = fmaxf(s_red[tid], s_red[tid + s]);
        __syncthreads();
    }
    const float maxv = s_red[0];

    // ---- load shifted/halved values, pad with -inf ----
    for (int idx = tid; idx < P; idx += ENT_THREADS)
        s_sorted[idx] = (idx < L) ? (z[idx] - maxv) * 0.5f : -__builtin_inff();
    __syncthreads();

    // ---- bitonic sort, descending ----
    for (int kk = 2; kk <= P; kk <<= 1) {
        for (int j = kk >> 1; j > 0; j >>= 1) {
            for (int idx = tid; idx < P; idx += ENT_THREADS) {
                const int ixj = idx ^ j;
                if (ixj > idx) {
                    const float a = s_sorted[idx];
                    const float b = s_sorted[ixj];
                    const bool desc = ((idx & kk) == 0);
                    if (desc ? (a < b) : (a > b)) {
                        s_sorted[idx] = b;
                        s_sorted[ixj] = a;
                    }
                }
            }
            __syncthreads();
        }
    }

    // ---- inclusive scans of v and v^2 (Hillis-Steele) ----
    for (int idx = tid; idx < P; idx += ENT_THREADS) {
        const float v = s_sorted[idx];
        s_cs[idx]  = v;
        s_cs2[idx] = v * v;
    }
    __syncthreads();
    for (int off = 1; off < P; off <<= 1) {
        float t1[4], t2[4];
        int c = 0;
        for (int idx = tid; idx < P; idx += ENT_THREADS, ++c) {
            t1[c] = (idx >= off) ? s_cs[idx - off]  : 0.0f;
            t2[c] = (idx >= off) ? s_cs2[idx - off] : 0.0f;
        }
        __syncthreads();
        c = 0;
        for (int idx = tid; idx < P; idx += ENT_THREADS, ++c) {
            s_cs[idx]  += t1[c];
            s_cs2[idx] += t2[c];
        }
        __syncthreads();
    }

    // ---- tau candidates + support count ----
    int mycnt = 0;
    for (int r = tid; r < P; r += ENT_THREADS) {
        const float rho     = (float)(r + 1);
        const float mean    = s_cs[r] / rho;
        const float mean_sq = s_cs2[r] / rho;
        const float ss      = rho * (mean_sq - mean * mean);
        const float delta   = (1.0f - ss) / rho;
        const float tau     = mean - sqrtf(fmaxf(delta, 0.0f));
        s_tau[r] = tau;
        // guard r < L: reproduces JAX NaN->False for -inf padded columns
        if (r < L && tau <= s_sorted[r]) ++mycnt;
    }
    if (mycnt) atomicAdd(&s_cnt, mycnt);
    __syncthreads();

    const int ksup = (s_cnt > 0) ? s_cnt : 1;
    const float tau_star = s_tau[ksup - 1];

    // ---- write full row: clip((z-max)/2 - tau, 0)^2, zeros past diagonal ----
    for (int j = tid; j < TT; j += ENT_THREADS) {
        float out = 0.0f;
        if (j < L) {
            const float t = (z[j] - maxv) * 0.5f - tau_star;
            out = (t > 0.0f) ? t * t : 0.0f;
        }
        z[j] = out;
    }
}

// ---------------------------------------------------------------------------
// Kernel 4: y_heads = att @ v   (per head: [T,T] x [T,hd]).
//  One wave per 16(m) x 64(full hd) strip: the att (A) fragment — the 201MB
//  tensor, the dominant HBM stream — is reused across all 4 V tiles.
//  Register double-buffered; K bounded by causality.
//  grid (T/16, B*H) = (64, 48), block 32.
// ---------------------------------------------------------------------------
__global__ void av_gemm_kernel(const float* __restrict__ att,
                               const float* __restrict__ v,
                               float* __restrict__ yh) {
    const int lane = threadIdx.x & 31;
    const int bh = blockIdx.y;
    const int m0 = blockIdx.x * 16;

    const float* A = att + (long)bh * TT * TT;
    const float* V = v   + (long)bh * TT * HD;

    const int l16  = lane & 15;
    const int koff = (lane < 16) ? 0 : 2;
    const int Kend = m0 + 16;                    // past-diagonal att is zero

    const float* pa = A + (long)(m0 + l16) * TT + koff;

    v8f acc[4] = {};
    // prime pipeline
    v2f a_cur = ldfrag(pa, 0);
    v2f b_cur[4];
    {
        const float* pv = V + (long)koff * HD + l16;
#pragma unroll
        for (int j = 0; j < 4; ++j) {
            b_cur[j].x = pv[j * 16];
            b_cur[j].y = pv[HD + j * 16];
        }
    }

    for (int kk = 0; kk + 4 < Kend; kk += 4) {
        __builtin_prefetch(pa + kk + 128, 0, 1); // global_prefetch_b8
        v2f a_nxt = ldfrag(pa, kk + 4);
        const float* pv = V + (long)(kk + 4 + koff) * HD + l16;
        v2f b_nxt[4];
#pragma unroll
        for (int j = 0; j < 4; ++j) {
            b_nxt[j].x = pv[j * 16];
            b_nxt[j].y = pv[HD + j * 16];
        }
#pragma unroll
        for (int j = 0; j < 4; ++j) acc[j] = wmma_f32(a_cur, b_cur[j], acc[j]);
        a_cur = a_nxt;
#pragma unroll
        for (int j = 0; j < 4; ++j) b_cur[j] = b_nxt[j];
    }
#pragma unroll
    for (int j = 0; j < 4; ++j) acc[j] = wmma_f32(a_cur, b_cur[j], acc[j]);

    const int rbase = m0 + ((lane < 16) ? 0 : 8);
#pragma unroll
    for (int j = 0; j < 4; ++j) {
        const int col = j * 16 + l16;
#pragma unroll
        for (int r = 0; r < 8; ++r)
            yh[((long)bh * TT + rbase + r) * HD + col] = acc[j][r];
    }
}

// ---------------------------------------------------------------------------
// Kernel 5: y = gather(y_heads) @ W_proj^T + b_proj -> d_out y region
//  M = 4096, N = 768, K = 768; A[m][k] gathered from [B,H,T,hd].
//  Block 128 = 4 waves, each a 16x64 strip; register double-buffered.
//  grid (256, 3).
// ---------------------------------------------------------------------------
__global__ void proj_gemm_kernel(const float* __restrict__ yh,
                                 const float* __restrict__ W,
                                 const float* __restrict__ bias,
                                 float* __restrict__ out) {
    const int lane = threadIdx.x & 31;
    const int wave = threadIdx.x >> 5;
    const int m0 = blockIdx.x * 16;
    const int n0 = (blockIdx.y * 4 + wave) * 64;

    const int l16  = lane & 15;
    const int koff = (lane < 16) ? 0 : 2;

    const int m    = m0 + l16;
    const int bidx = m >> 10;
    const int t    = m & (TT - 1);
    const float* pa_base = yh + (((long)bidx * HH) * TT + t) * HD;

    const float* pb[4];
#pragma unroll
    for (int j = 0; j < 4; ++j)
        pb[j] = W + (long)(n0 + j * 16 + l16) * CC + koff;

    // A gather for k-index kidx (even): head h = kidx>>6, dim d = kidx&63
    auto lda = [&](int kidx) -> v2f {
        const float* ap = pa_base + (long)(kidx >> 6) * TT * HD + (kidx & 63);
        v2f a; a.x = ap[0]; a.y = ap[1]; return a;
    };

    v8f acc[4] = {};
    v2f a_cur = lda(koff);
    v2f b_cur[4];
#pragma unroll
    for (int j = 0; j < 4; ++j) b_cur[j] = ldfrag(pb[j], 0);

    for (int kk = 0; kk + 4 < CC; kk += 4) {
        v2f a_nxt = lda(kk + 4 + koff);
        v2f b_nxt[4];
#pragma unroll
        for (int j = 0; j < 4; ++j) b_nxt[j] = ldfrag(pb[j], kk + 4);
#pragma unroll
        for (int j = 0; j < 4; ++j) acc[j] = wmma_f32(a_cur, b_cur[j], acc[j]);
        a_cur = a_nxt;
#pragma unroll
        for (int j = 0; j < 4; ++j) b_cur[j] = b_nxt[j];
    }
#pragma unroll
    for (int j = 0; j < 4; ++j) acc[j] = wmma_f32(a_cur, b_cur[j], acc[j]);

    const int rbase = m0 + ((lane < 16) ? 0 : 8);
#pragma unroll
    for (int j = 0; j < 4; ++j) {
        const int col  = n0 + j * 16 + l16;
        const float bv = bias[col];
#pragma unroll
        for (int r = 0; r < 8; ++r)
            out[(long)(rbase + r) * CC + col] = acc[j][r] + bv;
    }
}

// ---------------------------------------------------------------------------
extern "C" void kernel_launch(void* const* d_in, const int* in_sizes, int n_in,
                              void* d_out, int out_size, void* d_ws, size_t ws_size,
                              hipStream_t stream) {
    const float* x      = (const float*)d_in[0];
    const float* W_attn = (const float*)d_in[1];
    const float* b_attn = (const float*)d_in[2];
    const float* W_proj = (const float*)d_in[3];
    const float* b_proj = (const float*)d_in[4];

    float* y_out   = (float*)d_out;                       // [B,T,C]
    float* att_out = y_out + (long)BB * TT * CC;          // [B,H,T,T]

    const long NH = (long)BB * HH * TT * HD;              // per-tensor head elems
    float* q  = (float*)d_ws;
    float* k  = q + NH;
    float* v  = k + NH;
    float* yh = v + NH;

    // 1) qkv projection (f32 WMMA, 16x64 strips, double-buffered)
    qkv_gemm_kernel<<<dim3(BB * TT / 16, (3 * CC) / 256), 128, 0, stream>>>(
        x, W_attn, b_attn, q, k, v);

    // 2) causal scores into d_out att region (doubles as scratch)
    scores_kernel<<<dim3(TT / 16, TT / 64, BB * HH), 32, 0, stream>>>(
        q, k, att_out);

    // 3) entmax-1.5 per row, in place (also zeroes masked region)
    entmax15_kernel<<<dim3(BB * HH * TT), ENT_THREADS, 0, stream>>>(att_out);

    // 4) y_heads = att @ v  (att fragment reused across full head dim)
    av_gemm_kernel<<<dim3(TT / 16, BB * HH), 32, 0, stream>>>(att_out, v, yh);

    // 5) output projection
    proj_gemm_kernel<<<dim3(BB * TT / 16, CC / 256), 128, 0, stream>>>(
        yh, W_proj, b_proj, y_out);
}